// FernSparseTable_tabular_44779329028745
// MI455X (gfx1250) — compile-verified
//
#include <hip/hip_runtime.h>

// Problem constants (match reference)
#define K_BITS   10
#define M_FERNS  32
#define P_PAT    16
#define LPBITS   4
#define D_OUT    64
#define N_SAMP   4096
#define TABLE    1024
#define NBLK     64                     // TABLE / P_PAT
#define NBINS    (M_FERNS * NBLK)       // 2048 fast bins (+1 slow bucket)
#define NITEMS   (N_SAMP * M_FERNS)     // 131072

typedef __attribute__((ext_vector_type(2))) float v2f;
typedef __attribute__((ext_vector_type(8))) float v8f;

struct Prep {
  float AT[P_PAT];   // activations per pattern
  float s[LPBITS];   // selected soft-bit values
  int   abi[LPBITS]; // selected bit indices
  int   wb;          // thresholded word
  bool  fast;        // abi == {0,1,2,3}
};

// Faithful per-(n,m) reproduction of _activations_and_indices (prune_type==1).
__device__ __forceinline__ void fern_prep(const float* __restrict__ Bp,
                                          int n, int m, Prep& pr) {
  float T[K_BITS];
  const float* t = Bp + ((size_t)n * M_FERNS + m) * K_BITS;
#pragma unroll
  for (int j = 0; j < K_BITS; ++j) T[j] = t[j];

  int wb = 0; float probs = 1.0f;
#pragma unroll
  for (int j = 0; j < K_BITS; ++j) {
    bool b = T[j] >= 0.5f;
    wb |= ((int)b) << j;
    probs *= b ? T[j] : (1.0f - T[j]);
  }

  // Iterative ambiguous-bit selection: argmax of elig*(K-1-j), first-max wins.
  int helpmask = 0;
#pragma unroll
  for (int it = 0; it < LPBITS; ++it) {
    int best = 0; float bestScore = -1.0e30f; float sval = 0.0f;
#pragma unroll
    for (int j = 0; j < K_BITS; ++j) {
      bool helped = (helpmask >> j) & 1;
      float BAj   = helped ? 0.0f : T[j];
      float elig  = helped ? -1.0f
                  : ((BAj != 0.0f && BAj != 1.0f) ? 1.0f : 0.0f);
      float score = elig * (float)(K_BITS - 1 - j);
      if (score > bestScore) { bestScore = score; best = j; sval = BAj; }
    }
    pr.abi[it] = best; pr.s[it] = sval;
    helpmask |= 1 << best;
  }

  float denom = 1.0f;
#pragma unroll
  for (int it = 0; it < LPBITS; ++it)
    denom *= (pr.s[it] < 0.5f) ? (1.0f - pr.s[it]) : pr.s[it];
  float bsp = probs / denom;

#pragma unroll
  for (int p = 0; p < P_PAT; ++p) {
    float a = bsp;
#pragma unroll
    for (int j = 0; j < LPBITS; ++j)
      a *= ((p >> j) & 1) ? pr.s[j] : (1.0f - pr.s[j]);
    pr.AT[p] = a;
  }
  pr.wb = wb;
  pr.fast = (pr.abi[0] == 0) & (pr.abi[1] == 1) &
            (pr.abi[2] == 2) & (pr.abi[3] == 3);
}

__device__ __forceinline__ int item_bin(const Prep& pr, int m) {
  return pr.fast ? (m * NBLK + (pr.wb >> LPBITS)) : NBINS;
}

__global__ void k_zero(int* __restrict__ cnt) {
  int i = blockIdx.x * blockDim.x + threadIdx.x;
  if (i < NBINS + 1) cnt[i] = 0;
}

__global__ void k_init_out(float* __restrict__ out, const float* __restrict__ bias) {
  int i = blockIdx.x * blockDim.x + threadIdx.x;
  if (i < N_SAMP * D_OUT) out[i] = bias[i & (D_OUT - 1)];
}

__global__ void k_count(const float* __restrict__ B, int* __restrict__ cnt) {
  int tid = blockIdx.x * blockDim.x + threadIdx.x;
  if (tid >= NITEMS) return;
  int n = tid >> 5, m = tid & 31;
  Prep pr; fern_prep(B, n, m, pr);
  atomicAdd(&cnt[item_bin(pr, m)], 1);
}

__global__ void k_scan(const int* __restrict__ cnt,
                       int* __restrict__ off, int* __restrict__ cur) {
  if (blockIdx.x == 0 && threadIdx.x == 0) {
    int acc = 0;
    for (int i = 0; i <= NBINS; ++i) { off[i] = acc; cur[i] = acc; acc += cnt[i]; }
  }
}

__global__ void k_scatter(const float* __restrict__ B,
                          int* __restrict__ cur, int* __restrict__ items) {
  int tid = blockIdx.x * blockDim.x + threadIdx.x;
  if (tid >= NITEMS) return;
  int n = tid >> 5, m = tid & 31;
  Prep pr; fern_prep(B, n, m, pr);
  int pos = atomicAdd(&cur[item_bin(pr, m)], 1);
  items[pos] = n | (m << 16);
}

// One wave per (fern, 16-row block) bin. All items in a bin share the same
// 16x64 weight block -> dense fp32 WMMA: D(16 samp x 16 dim) += A(16x16 AT)
// x B(16 pat x 16 dim), chained as 4x V_WMMA_F32_16X16X4_F32 per dim-chunk.
__global__ void __launch_bounds__(32)
k_wmma(const float* __restrict__ B, const float* __restrict__ W,
       const int* __restrict__ cnt, const int* __restrict__ off,
       const int* __restrict__ items, float* __restrict__ out) {
  const int bin = blockIdx.x;
  const int count = cnt[bin];
  if (count == 0) return;
  const int m   = bin >> 6;
  const int blk = bin & 63;
  const int start = off[bin];
  const int lane = threadIdx.x;
  const int r = lane & 15;   // row (A: sample, B/D: dim column)
  const int h = lane >> 4;   // K half:  K = 2*h + vgpr

  // Cache the 16x64 block in registers, laid out for WMMA B-matrix:
  // bv[dc][t].{x,y} = W[base + 4t + 2h + {0,1}][dc*16 + r]
  const float* Wm = W + ((size_t)m * TABLE + (size_t)blk * P_PAT) * D_OUT;
  v2f bv[4][4];
#pragma unroll
  for (int t = 0; t < 4; ++t) {
#pragma unroll
    for (int dc = 0; dc < 4; ++dc) {
      const float* p0 = Wm + (4 * t + 2 * h) * D_OUT + dc * 16 + r;
      bv[dc][t].x = p0[0];
      bv[dc][t].y = p0[D_OUT];
    }
  }

  for (int c0 = 0; c0 < count; c0 += 16) {
    const int  idx   = c0 + r;
    const bool valid = idx < count;
    const int  n_i   = valid ? (items[start + idx] & 0xffff) : 0;

    Prep pr; fern_prep(B, n_i, m, pr);

    v8f acc[4] = {};  // 4 dim-chunks of D (16x16 f32 each)
#pragma unroll
    for (int t = 0; t < 4; ++t) {
      v2f a;  // A-matrix: lane holds AT[4t + 2h + {0,1}] of its sample row
      a.x = h ? pr.AT[4 * t + 2] : pr.AT[4 * t + 0];
      a.y = h ? pr.AT[4 * t + 3] : pr.AT[4 * t + 1];
      if (!valid) { a.x = 0.0f; a.y = 0.0f; }
#pragma unroll
      for (int dc = 0; dc < 4; ++dc) {
        acc[dc] = __builtin_amdgcn_wmma_f32_16x16x4_f32(
            false, a, false, bv[dc][t], (short)0, acc[dc], false, false);
      }
    }

    // D layout: vgpr j, lane l -> sample row M = j + 8*(l>>4), dim N = l&15.
    // Fetch each D-row's sample id from the lane that owns it, then scatter.
#pragma unroll
    for (int j = 0; j < 8; ++j) {
      const int srcLane = j + 8 * h;
      const int n_row = __shfl(n_i, srcLane, 32);
      if (c0 + srcLane < count) {
#pragma unroll
        for (int dc = 0; dc < 4; ++dc) {
          __hip_atomic_fetch_add(&out[(size_t)n_row * D_OUT + dc * 16 + r],
                                 acc[dc][j], __ATOMIC_RELAXED,
                                 __HIP_MEMORY_SCOPE_AGENT);
        }
      }
    }
  }
}

// Scalar fallback for items whose ambiguous bits are not {0,1,2,3}
// (possible only when an input bit is exactly 0.0/1.0). One wave per item.
__global__ void __launch_bounds__(32)
k_slow(const float* __restrict__ B, const float* __restrict__ W,
       const int* __restrict__ cnt, const int* __restrict__ off,
       const int* __restrict__ items, float* __restrict__ out) {
  const int lane = threadIdx.x;
  const int nSlow = cnt[NBINS];
  const int start = off[NBINS];
  for (int i = blockIdx.x; i < nSlow; i += gridDim.x) {
    const int code = items[start + i];
    const int n = code & 0xffff, m = code >> 16;
    Prep pr; fern_prep(B, n, m, pr);
    float acc0 = 0.0f, acc1 = 0.0f;
    const int full = TABLE - 1;
#pragma unroll
    for (int p = 0; p < P_PAT; ++p) {
      int it = pr.wb;
#pragma unroll
      for (int j = 0; j < LPBITS; ++j) {
        int mask = 1 << pr.abi[j];
        it = ((p >> j) & 1) ? (it | mask) : (it & (mask ^ full));
      }
      const float* row = W + ((size_t)m * TABLE + it) * D_OUT;
      acc0 += pr.AT[p] * row[2 * lane + 0];
      acc1 += pr.AT[p] * row[2 * lane + 1];
    }
    __hip_atomic_fetch_add(&out[(size_t)n * D_OUT + 2 * lane + 0], acc0,
                           __ATOMIC_RELAXED, __HIP_MEMORY_SCOPE_AGENT);
    __hip_atomic_fetch_add(&out[(size_t)n * D_OUT + 2 * lane + 1], acc1,
                           __ATOMIC_RELAXED, __HIP_MEMORY_SCOPE_AGENT);
  }
}

extern "C" void kernel_launch(void* const* d_in, const int* in_sizes, int n_in,
                              void* d_out, int out_size, void* d_ws, size_t ws_size,
                              hipStream_t stream) {
  const float* B    = (const float*)d_in[0];   // (N, M, K)
  const float* W    = (const float*)d_in[1];   // (M, TABLE, D_OUT)
  const float* bias = (const float*)d_in[2];   // (D_OUT,)
  float* out = (float*)d_out;                  // (N, D_OUT)

  int* cnt   = (int*)d_ws;          // NBINS+1 counts
  int* off   = cnt + (NBINS + 1);   // NBINS+1 start offsets
  int* cur   = off + (NBINS + 1);   // NBINS+1 scatter cursors
  int* items = cur + (NBINS + 1);   // NITEMS sorted item codes
  (void)in_sizes; (void)n_in; (void)out_size; (void)ws_size;

  hipLaunchKernelGGL(k_zero, dim3((NBINS + 1 + 255) / 256), dim3(256), 0, stream, cnt);
  hipLaunchKernelGGL(k_init_out, dim3((N_SAMP * D_OUT) / 256), dim3(256), 0, stream, out, bias);
  hipLaunchKernelGGL(k_count, dim3(NITEMS / 256), dim3(256), 0, stream, B, cnt);
  hipLaunchKernelGGL(k_scan, dim3(1), dim3(1), 0, stream, cnt, off, cur);
  hipLaunchKernelGGL(k_scatter, dim3(NITEMS / 256), dim3(256), 0, stream, B, cur, items);
  hipLaunchKernelGGL(k_wmma, dim3(NBINS), dim3(32), 0, stream, B, W, cnt, off, items, out);
  hipLaunchKernelGGL(k_slow, dim3(1024), dim3(32), 0, stream, B, W, cnt, off, items, out);
}